// Net_91113436217372
// MI455X (gfx1250) — compile-verified
//
#include <hip/hip_runtime.h>
#include <hip/hip_bf16.h>
#include <stdint.h>

// ---------------------------------------------------------------------------
// MI455X (gfx1250).
//   convs: v_wmma_f32_16x16x4_f32 (M=16 px, N=16 (3 oc), K=28), A staged in LDS
//          via TDM tensor_load_to_lds for interior tiles (s_wait_tensorcnt),
//          generic path for border tiles.
//   pools: elementwise.  fc1/fc2: BW-bound GEMV + global_prefetch_b8.
// ---------------------------------------------------------------------------

typedef __attribute__((ext_vector_type(2))) float v2f;
typedef __attribute__((ext_vector_type(8))) float v8f;
typedef __attribute__((ext_vector_type(4))) unsigned int v4u;
typedef __attribute__((ext_vector_type(8))) int v8i_;
typedef __attribute__((ext_vector_type(4))) int v4i_;

#define TH 16
#define TW 64
#define TCOLS (TW + 2)        // 66
#define TROWS (TH + 2)        // 18
#define PLANE (TROWS * TCOLS) // 1188

__global__ __launch_bounds__(256)
void conv3_wmma(const float* __restrict__ in, const float* __restrict__ w,
                float* __restrict__ out, int H, int W)
{
    // plane 3 is a zero plane: K-dim padding tap (k==27) reads from it, branch-free
    __shared__ float tile[4][TROWS][TCOLS];
    __shared__ float wk[28][16];   // padded B-matrix source wk[k][n]

    const int bx  = blockIdx.x * TW;
    const int by  = blockIdx.y * TH;
    const int tid = threadIdx.x;

    // ---- stage weights: tap k=(ky*3+kx)*3+c ; OIHW flat ----
    for (int i = tid; i < 28 * 16; i += 256) {
        int k = i >> 4, n = i & 15;
        float val = 0.0f;
        if (k < 27 && n < 3) {
            int c = k % 3, q = k / 3, kx = q % 3, ky = q / 3;
            val = w[((n * 3 + c) * 3 + ky) * 3 + kx];
        }
        wk[k][n] = val;
    }
    // ---- zero plane for K padding ----
    for (int i = tid; i < PLANE; i += 256) (&tile[3][0][0])[i] = 0.0f;

    const bool interior = (bx > 0) && (by > 0) && (bx + TW < W) && (by + TH < H);
    if (interior) {
        // ---- Tensor Data Mover: 3D tile (66 x 18 x 3ch) -> LDS, contiguous ----
        if (tid < 32) {  // wave 0 only (uniform branch)
            uint64_t ga = (uint64_t)(uintptr_t)in +
                          (uint64_t)(((int64_t)(by - 1) * W + (bx - 1)) * 4);
            uint32_t lds = (uint32_t)(uintptr_t)&tile[0][0][0];

            v4u g0;
            g0[0] = 1u;                                   // count=1, user desc
            g0[1] = lds;                                  // lds_addr (bytes)
            g0[2] = (uint32_t)(ga & 0xffffffffu);         // global_addr[31:0]
            g0[3] = (uint32_t)((ga >> 32) & 0x01ffffffu)  // global_addr[56:32]
                  | (2u << 30);                           // type=2 ("image")

            const uint32_t dim0 = (uint32_t)W;            // x bound
            const uint32_t dim1 = (uint32_t)H;            // y bound
            const uint64_t s0   = (uint64_t)W;            // row stride (elems)
            const uint64_t s1   = (uint64_t)H * (uint64_t)W; // plane stride

            v8i_ g1;
            g1[0] = (int)(2u << 16);                                   // data_size=4B
            g1[1] = (int)((dim0 & 0xffffu) << 16);                     // dim0[15:0]
            g1[2] = (int)((dim0 >> 16) | ((dim1 & 0xffffu) << 16));    // dim0[31:16],dim1[15:0]
            g1[3] = (int)((dim1 >> 16) | ((uint32_t)TCOLS << 16));     // dim1[31:16],tile_dim0=66
            g1[4] = (int)((uint32_t)TROWS | (3u << 16));               // tile_dim1=18,tile_dim2=3
            g1[5] = (int)(uint32_t)(s0 & 0xffffffffu);                 // dim0_stride[31:0]
            g1[6] = (int)(uint32_t)(((s0 >> 32) & 0xffffu) |
                                    ((s1 & 0xffffu) << 16));           // dim0_str[47:32],dim1_str[15:0]
            g1[7] = (int)(uint32_t)((s1 >> 16) & 0xffffffffu);         // dim1_stride[47:16]

            v4i_ g2; g2[0] = 3; g2[1] = 0; g2[2] = 0; g2[3] = 0;       // tensor_dim2=3
            v4i_ g3 = {0, 0, 0, 0};

#if __clang_major__ >= 23
            v8i_ g4 = {0, 0, 0, 0, 0, 0, 0, 0};
            __builtin_amdgcn_tensor_load_to_lds(g0, g1, g2, g3, g4, 0);
#else
            __builtin_amdgcn_tensor_load_to_lds(g0, g1, g2, g3, 0);
#endif
            __builtin_amdgcn_s_wait_tensorcnt(0);
        }
    } else {
        // ---- border tiles: generic staging with zero padding ----
        for (int i = tid; i < 3 * PLANE; i += 256) {
            int c   = i / PLANE;
            int rem = i % PLANE;
            int r   = rem / TCOLS;
            int col = rem % TCOLS;
            int gy = by + r - 1, gx = bx + col - 1;
            float v = 0.0f;
            if (gy >= 0 && gy < H && gx >= 0 && gx < W)
                v = in[(c * H + gy) * W + gx];
            tile[c][r][col] = v;
        }
    }
    __syncthreads();

    const int lane = tid & 31;
    const int m    = lane & 15;   // A: M pixel idx / B: N column idx
    const int h    = lane >> 4;   // lane half -> K pair {2h,2h+1}
    const int wave = tid >> 5;

    // per-lane strip-invariant tap offsets (element units); tap 27 -> zero plane
    int offe[14];
#pragma unroll
    for (int t = 0; t < 7; ++t)
#pragma unroll
        for (int vv = 0; vv < 2; ++vv) {
            int k = 4 * t + 2 * h + vv;
            int c, ky, kx;
            if (k >= 27) { c = 3; ky = 0; kx = 0; }
            else { c = k % 3; int q = k / 3; kx = q % 3; ky = q / 3; }
            offe[2 * t + vv] = (c * TROWS + ky) * TCOLS + kx;
        }

    // B operand held in VGPRs for the whole tile
    v2f bk[7];
#pragma unroll
    for (int t = 0; t < 7; ++t) {
        int k0 = 4 * t + 2 * h;
        bk[t].x = wk[k0 + 0][m];
        bk[t].y = wk[k0 + 1][m];
    }

    const float* tf = &tile[0][0][0];

    // 64 strips (16 rows x 4 col groups), uniform control flow around WMMA
    for (int s = wave; s < TH * (TW / 16); s += 8) {
        int r    = s >> 2;
        int xc   = (s & 3) << 4;
        int base = r * TCOLS + xc + m;

        // gather all 14 A-taps first -> single dscnt wait, then 7 WMMAs b2b
        float av[14];
#pragma unroll
        for (int i = 0; i < 14; ++i) av[i] = tf[base + offe[i]];

        v8f acc = {0.f, 0.f, 0.f, 0.f, 0.f, 0.f, 0.f, 0.f};
#pragma unroll
        for (int t = 0; t < 7; ++t) {
            v2f a; a.x = av[2 * t]; a.y = av[2 * t + 1];
            acc = __builtin_amdgcn_wmma_f32_16x16x4_f32(
                false, a, false, bk[t], (short)0, acc, false, false);
        }

        // D: vgpr v -> M = v + 8*h, lane%16 -> N (output channel)
        if (m < 3) {
            int gy  = by + r;
            int gx0 = bx + xc + h * 8;
            float* o = out + ((m * H) + gy) * W + gx0;
#pragma unroll
            for (int vv = 0; vv < 8; ++vv) o[vv] = acc[vv];
        }
    }
}

// 2x2 max pool, stride 2, C=3.
__global__ __launch_bounds__(256)
void maxpool2k(const float* __restrict__ in, float* __restrict__ out, int H, int W)
{
    int idx = blockIdx.x * blockDim.x + threadIdx.x;
    int oW = W >> 1, oH = H >> 1;
    int total = 3 * oH * oW;
    if (idx >= total) return;
    int ox = idx % oW;
    int t  = idx / oW;
    int oy = t % oH;
    int c  = t / oH;
    const float* p = in + ((size_t)c * H + 2 * oy) * W + 2 * ox;
    float m0 = fmaxf(p[0], p[1]);
    float m1 = fmaxf(p[W], p[W + 1]);
    out[idx] = fmaxf(m0, m1);
}

// out[j] = dot(Wm[j,:], v), one block per row; HBM-bandwidth bound.
__global__ __launch_bounds__(256)
void fc_matvec(const float* __restrict__ Wm, const float* __restrict__ v,
               float* __restrict__ out, int K)
{
    __shared__ float red[8];
    const int j   = blockIdx.x;
    const int tid = threadIdx.x;
    const float4* row4 = (const float4*)(Wm + (size_t)j * K);
    const float4* v4   = (const float4*)v;
    const int K4 = K >> 2;

    float sum = 0.0f;
    for (int i = tid; i < K4; i += 256) {
        __builtin_prefetch((const void*)(row4 + i + 512), 0, 0);
        float4 a = row4[i];
        float4 b = v4[i];
        sum += a.x * b.x + a.y * b.y + a.z * b.z + a.w * b.w;
    }
    for (int off = 16; off > 0; off >>= 1) sum += __shfl_down(sum, off, 32);
    if ((tid & 31) == 0) red[tid >> 5] = sum;
    __syncthreads();
    if (tid < 8) {
        float s = red[tid];
        for (int off = 4; off > 0; off >>= 1) s += __shfl_down(s, off, 8);
        if (tid == 0) out[j] = s;
    }
}

extern "C" void kernel_launch(void* const* d_in, const int* in_sizes, int n_in,
                              void* d_out, int out_size, void* d_ws, size_t ws_size,
                              hipStream_t stream)
{
    (void)in_sizes; (void)n_in; (void)out_size; (void)ws_size;

    const float* x = (const float*)d_in[0];
    const float* w[13];
    for (int i = 0; i < 13; ++i) w[i] = (const float*)d_in[5 + i];
    const float* fc1 = (const float*)d_in[18];
    const float* fc2 = (const float*)d_in[19];
    float* outp = (float*)d_out;

    const size_t IMG = ((size_t)3 * 5120 * 5120 * sizeof(float) + 255) & ~(size_t)255;
    float* A = (float*)d_ws;
    float* B = (float*)((char*)d_ws + IMG);

    auto conv = [&](const float* i, const float* wt, float* o, int HW) {
        dim3 g(HW / TW, HW / TH);
        conv3_wmma<<<g, 256, 0, stream>>>(i, wt, o, HW, HW);
    };
    auto pool = [&](const float* i, float* o, int HW) {
        int total = 3 * (HW / 2) * (HW / 2);
        maxpool2k<<<(total + 255) / 256, 256, 0, stream>>>(i, o, HW, HW);
    };

    conv(x, w[0], A, 5120);  conv(A, w[1], B, 5120);  pool(B, A, 5120);
    conv(A, w[2], B, 2560);  conv(B, w[3], A, 2560);  pool(A, B, 2560);
    conv(B, w[4], A, 1280);  conv(A, w[5], B, 1280);  conv(B, w[6], A, 1280);  pool(A, B, 1280);
    conv(B, w[7], A, 640);   conv(A, w[8], B, 640);   conv(B, w[9], A, 640);   pool(A, B, 640);
    conv(B, w[10], A, 320);  conv(A, w[11], B, 320);  conv(B, w[12], A, 320);  pool(A, B, 320);

    fc_matvec<<<1024, 256, 0, stream>>>(fc1, B, A, 76800);
    fc_matvec<<<1024, 256, 0, stream>>>(fc2, A, outp, 1024);
}